// MambaOnlyBlock_63634235458202
// MI455X (gfx1250) — compile-verified
//
#include <hip/hip_runtime.h>
#include <hip/hip_bf16.h>
#include <math.h>

typedef float v2f __attribute__((ext_vector_type(2)));
typedef float v8f __attribute__((ext_vector_type(8)));

namespace {
constexpr int Bb   = 2;      // batch
constexpr int NC   = 64;     // channels C
constexpr int Nn   = 16384;  // H*W tokens per batch (2^14)
constexpr int NT   = Bb * Nn;
constexpr int DI   = 128;    // d_inner (2^7)
constexpr int DS   = 16;     // d_state
constexpr int CHUNK = 64;
constexpr int NCH  = Nn / CHUNK; // 256 chunks per batch
}

static __device__ __forceinline__ float silu_f(float v) {
    return v / (1.0f + __expf(-v));
}

static __device__ __forceinline__ v8f wmma_f32(v2f a, v2f b, v8f c) {
    // V_WMMA_F32_16X16X4_F32 : D(16x16,f32) = A(16x4,f32) * B(4x16,f32) + C
    return __builtin_amdgcn_wmma_f32_16x16x4_f32(false, a, false, b, (short)0, c,
                                                 false, false);
}

// -------------------------------------------------------------------------
// Kernel 1: LayerNorm(C=64) + in_proj (64 -> 256), 64 tokens per block.
// -------------------------------------------------------------------------
__global__ __launch_bounds__(256) void k_ln_inproj(
    const float* __restrict__ x, const float* __restrict__ lnw_g,
    const float* __restrict__ lnb_g, const float* __restrict__ w,
    float* __restrict__ xin, float* __restrict__ zb)
{
    __shared__ float tmat[64 * 65];  // 64 tokens x 64 feats, pad 65
    __shared__ float lnw[64];
    __shared__ float lnb[64];
    __shared__ float ps1[256];
    __shared__ float ps2[256];
    __shared__ float muA[64];
    __shared__ float rsA[64];

    const int tid = threadIdx.x;
    const int g0  = blockIdx.x * 64;       // first global token of tile
    const int b   = g0 >> 14;              // g0 / Nn
    const int n0  = g0 & (Nn - 1);

    if (tid < 64) { lnw[tid] = lnw_g[tid]; lnb[tid] = lnb_g[tid]; }

    // Load x tile (B,C,H*W layout) -> transposed LDS [token][channel].
    // Each thread: 16 consecutive floats (64B-aligned) as 4x float4.
    {
        const int c  = tid >> 2;
        const int jb = (tid & 3) << 4;
        const float4* xp4 = reinterpret_cast<const float4*>(
            x + (size_t)b * NC * Nn + (size_t)c * Nn + n0 + jb);
        #pragma unroll
        for (int j4 = 0; j4 < 4; ++j4) {
            const float4 v4 = xp4[j4];
            tmat[(jb + 4 * j4 + 0) * 65 + c] = v4.x;
            tmat[(jb + 4 * j4 + 1) * 65 + c] = v4.y;
            tmat[(jb + 4 * j4 + 2) * 65 + c] = v4.z;
            tmat[(jb + 4 * j4 + 3) * 65 + c] = v4.w;
        }
    }
    __syncthreads();

    // LayerNorm: 4 threads per token, 16-channel partials -> combine.
    {
        const int j = tid >> 2, q = tid & 3;
        float s1 = 0.f, s2 = 0.f;
        #pragma unroll
        for (int cc = 0; cc < 16; ++cc) {
            const float v = tmat[j * 65 + q * 16 + cc];
            s1 += v; s2 += v * v;
        }
        ps1[tid] = s1; ps2[tid] = s2;
    }
    __syncthreads();
    if (tid < 64) {
        const float s1 = ps1[tid * 4] + ps1[tid * 4 + 1] + ps1[tid * 4 + 2] + ps1[tid * 4 + 3];
        const float s2 = ps2[tid * 4] + ps2[tid * 4 + 1] + ps2[tid * 4 + 2] + ps2[tid * 4 + 3];
        const float mu  = s1 * (1.0f / 64.0f);
        const float var = s2 * (1.0f / 64.0f) - mu * mu;
        muA[tid] = mu;
        rsA[tid] = rsqrtf(var + 1e-5f);
    }
    __syncthreads();
    {
        const int j = tid >> 2, q = tid & 3;
        const float mu = muA[j], rs = rsA[j];
        #pragma unroll
        for (int cc = 0; cc < 16; ++cc) {
            const int c2 = q * 16 + cc;
            const float v = tmat[j * 65 + c2];
            tmat[j * 65 + c2] = (v - mu) * rs * lnw[c2] + lnb[c2];
        }
    }
    __syncthreads();

    // GEMM: 4 token tiles x 16 output tiles = 64 jobs across 8 waves.
    const int wave = tid >> 5, lane = tid & 31;
    const int mrow = lane & 15, hi = lane >> 4;
    const int khalf = hi * 2;
    for (int i = 0; i < 8; ++i) {
        const int job = wave * 8 + i;
        const int tt = job >> 4, ot = job & 15;
        const float* arow = &tmat[(tt * 16 + mrow) * 65];
        const float* brow = w + (ot * 16 + mrow) * 64;   // [out][in] row-major
        v8f acc = {0.f, 0.f, 0.f, 0.f, 0.f, 0.f, 0.f, 0.f};
        #pragma unroll
        for (int kk = 0; kk < 16; ++kk) {
            const int k0 = kk * 4 + khalf;
            v2f av = { arow[k0], arow[k0 + 1] };
            v2f bv = { brow[k0], brow[k0 + 1] };
            acc = wmma_f32(av, bv, acc);
        }
        // Destination is uniform per job (ot<8 -> xin, else z): no divergence.
        const int oi = ot * 16 + mrow;
        float* dstBase = (ot < 8)
            ? (xin + (size_t)(g0 + tt * 16 + hi * 8) * DI + oi)
            : (zb  + (size_t)(g0 + tt * 16 + hi * 8) * DI + (oi - DI));
        #pragma unroll
        for (int r = 0; r < 8; ++r)
            dstBase[(size_t)r * DI] = acc[r];
    }
}

// -------------------------------------------------------------------------
// Kernel 2: depthwise causal conv1d (k=4, left pad 3) + SiLU.
// -------------------------------------------------------------------------
__global__ __launch_bounds__(256) void k_conv_silu(
    const float* __restrict__ xin, const float* __restrict__ cw,
    const float* __restrict__ cb, float* __restrict__ u)
{
    const int idx = blockIdx.x * 256 + threadIdx.x;   // < NT*DI
    const int d = idx & (DI - 1);
    const int g = idx >> 7;
    const int b = g >> 14;
    const int n = g & (Nn - 1);
    float acc = cb[d];
    #pragma unroll
    for (int k = 0; k < 4; ++k) {
        const int m = n + k - 3;
        if (m >= 0)
            acc += cw[d * 4 + k] * xin[(size_t)(b * Nn + m) * DI + d];
    }
    u[(size_t)g * DI + d] = silu_f(acc);
}

// -------------------------------------------------------------------------
// Kernel 3: x_proj (128 -> 36, WMMA padded to 48) + dt = softplus(dt_proj).
// -------------------------------------------------------------------------
__global__ __launch_bounds__(128) void k_xproj_dt(
    const float* __restrict__ u, const float* __restrict__ xpw,
    const float* __restrict__ dtw, const float* __restrict__ dtb,
    float* __restrict__ dt, float* __restrict__ Bco, float* __restrict__ Cco)
{
    __shared__ float ulds[64 * 132];  // 64 tokens x 128, pad 132
    __shared__ float xdl[64 * 49];    // 64 tokens x 48 (x_dbl), pad 49

    const int tid = threadIdx.x;
    const int g0  = blockIdx.x * 64;

    for (int i = tid; i < 64 * DI; i += 128)
        ulds[(i >> 7) * 132 + (i & 127)] = u[(size_t)g0 * DI + i];
    __syncthreads();

    const int wave = tid >> 5, lane = tid & 31;
    const int mrow = lane & 15, hi = lane >> 4;
    const int khalf = hi * 2;
    for (int i = 0; i < 3; ++i) {
        const int job = wave * 3 + i;     // 12 jobs: tt(0..3) x ot(0..2)
        const int tt = job / 3, ot = job % 3;
        const int row = ot * 16 + mrow;   // output row, valid < 36
        const bool valid = row < 36;
        const float* arow = &ulds[(tt * 16 + mrow) * 132];
        const float* brow = xpw + (valid ? row : 0) * 128;
        v8f acc = {0.f, 0.f, 0.f, 0.f, 0.f, 0.f, 0.f, 0.f};
        #pragma unroll
        for (int kk = 0; kk < 32; ++kk) {
            const int k0 = kk * 4 + khalf;
            v2f av = { arow[k0], arow[k0 + 1] };
            const float b0 = valid ? brow[k0] : 0.f;
            const float b1 = valid ? brow[k0 + 1] : 0.f;
            v2f bv = { b0, b1 };
            acc = wmma_f32(av, bv, acc);
        }
        #pragma unroll
        for (int r = 0; r < 8; ++r)
            xdl[(tt * 16 + r + hi * 8) * 49 + ot * 16 + mrow] = acc[r];
    }
    __syncthreads();

    // dt = softplus(x_dbl[0:4] @ dt_proj_w^T + b); one channel per thread.
    {
        const float w0 = dtw[tid * 4 + 0], w1 = dtw[tid * 4 + 1];
        const float w2 = dtw[tid * 4 + 2], w3 = dtw[tid * 4 + 3];
        const float bb = dtb[tid];
        for (int j = 0; j < 64; ++j) {
            const float* xr = &xdl[j * 49];
            const float pre = bb + xr[0] * w0 + xr[1] * w1 + xr[2] * w2 + xr[3] * w3;
            const float sp = (pre > 20.f) ? pre : log1pf(expf(pre));
            dt[(size_t)(g0 + j) * DI + tid] = sp;
        }
    }
    // Split out B (x_dbl[4:20]) and C (x_dbl[20:36]).
    for (int idx = tid; idx < 64 * 32; idx += 128) {
        const int j = idx >> 5;
        const int e = idx & 31;
        const float v = xdl[j * 49 + 4 + e];
        const int tok = g0 + j;
        if (e < 16) Bco[tok * DS + e] = v;
        else        Cco[tok * DS + (e - 16)] = v;
    }
}

// -------------------------------------------------------------------------
// Kernel 4a: chunked scan, pass 1 -> per-chunk decay product + local state.
// B tile staged in LDS (coalesced once), broadcast ds reads in hot loop.
// -------------------------------------------------------------------------
__global__ __launch_bounds__(128) void k_scan_pass1(
    const float* __restrict__ dt, const float* __restrict__ u,
    const float* __restrict__ Bco, const float* __restrict__ Alog,
    float* __restrict__ prodA, float* __restrict__ hfin)
{
    __shared__ float Bs[CHUNK * DS];   // 4 KB

    const int d  = threadIdx.x;
    const int b  = blockIdx.x / NCH;
    const int ch = blockIdx.x % NCH;
    const int gbase = b * Nn + ch * CHUNK;

    for (int i = threadIdx.x; i < CHUNK * DS; i += 128)
        Bs[i] = Bco[(size_t)gbase * DS + i];

    float Ac[DS], h[DS], p[DS];
    #pragma unroll
    for (int s = 0; s < DS; ++s) {
        Ac[s] = -expf(Alog[d * DS + s]);
        h[s] = 0.f; p[s] = 1.f;
    }
    __syncthreads();

    const float* dtp = dt + (size_t)gbase * DI + d;
    const float* up  = u  + (size_t)gbase * DI + d;
    for (int i = 0; i < CHUNK; ++i) {
        const float dtv = dtp[(size_t)i * DI];
        const float uv  = up[(size_t)i * DI];
        const float du  = dtv * uv;
        const float4* Bp = reinterpret_cast<const float4*>(&Bs[i * DS]);
        float Bv[DS];
        #pragma unroll
        for (int q = 0; q < 4; ++q) {
            const float4 t4 = Bp[q];
            Bv[4 * q] = t4.x; Bv[4 * q + 1] = t4.y;
            Bv[4 * q + 2] = t4.z; Bv[4 * q + 3] = t4.w;
        }
        #pragma unroll
        for (int s = 0; s < DS; ++s) {
            const float dA = __expf(dtv * Ac[s]);
            h[s] = dA * h[s] + du * Bv[s];
            p[s] *= dA;
        }
    }
    const size_t base = ((size_t)(b * NCH + ch) * DI + d) * DS;
    #pragma unroll
    for (int s = 0; s < DS; ++s) { prodA[base + s] = p[s]; hfin[base + s] = h[s]; }
}

// -------------------------------------------------------------------------
// Kernel 4b: sequential combine of chunk summaries -> per-chunk carry-in.
// -------------------------------------------------------------------------
__global__ __launch_bounds__(256) void k_scan_carry(
    const float* __restrict__ prodA, const float* __restrict__ hfin,
    float* __restrict__ carry)
{
    const int t   = blockIdx.x * 256 + threadIdx.x;   // < Bb*DI*DS = 4096
    const int b   = t / (DI * DS);
    const int rem = t % (DI * DS);                    // d*DS + s
    float hacc = 0.f;
    for (int ch = 0; ch < NCH; ++ch) {
        const size_t base = (size_t)(b * NCH + ch) * DI * DS + rem;
        carry[base] = hacc;
        hacc = prodA[base] * hacc + hfin[base];
    }
}

// -------------------------------------------------------------------------
// Kernel 4c: scan pass 2 with carry-in; fuse y = h.C + u*D, gate by silu(z).
// B and C tiles staged in LDS.
// -------------------------------------------------------------------------
__global__ __launch_bounds__(128) void k_scan_pass2(
    const float* __restrict__ dt, const float* __restrict__ u,
    const float* __restrict__ zb, const float* __restrict__ Bco,
    const float* __restrict__ Cco, const float* __restrict__ Alog,
    const float* __restrict__ Dp, const float* __restrict__ carry,
    float* __restrict__ y)
{
    __shared__ float Bs[CHUNK * DS];   // 4 KB
    __shared__ float Cs[CHUNK * DS];   // 4 KB

    const int d  = threadIdx.x;
    const int b  = blockIdx.x / NCH;
    const int ch = blockIdx.x % NCH;
    const int gbase = b * Nn + ch * CHUNK;

    for (int i = threadIdx.x; i < CHUNK * DS; i += 128) {
        Bs[i] = Bco[(size_t)gbase * DS + i];
        Cs[i] = Cco[(size_t)gbase * DS + i];
    }

    const size_t cbase = ((size_t)(b * NCH + ch) * DI + d) * DS;
    float Ac[DS], h[DS];
    #pragma unroll
    for (int s = 0; s < DS; ++s) {
        Ac[s] = -expf(Alog[d * DS + s]);
        h[s] = carry[cbase + s];
    }
    const float Dv = Dp[d];
    __syncthreads();

    const float* dtp = dt + (size_t)gbase * DI + d;
    const float* up  = u  + (size_t)gbase * DI + d;
    const float* zp  = zb + (size_t)gbase * DI + d;
    float*       yp  = y  + (size_t)gbase * DI + d;
    for (int i = 0; i < CHUNK; ++i) {
        const float dtv = dtp[(size_t)i * DI];
        const float uv  = up[(size_t)i * DI];
        const float zv  = zp[(size_t)i * DI];
        const float du  = dtv * uv;
        const float4* Bp = reinterpret_cast<const float4*>(&Bs[i * DS]);
        const float4* Cp = reinterpret_cast<const float4*>(&Cs[i * DS]);
        float Bv[DS], Cv[DS];
        #pragma unroll
        for (int q = 0; q < 4; ++q) {
            const float4 t4 = Bp[q];
            Bv[4 * q] = t4.x; Bv[4 * q + 1] = t4.y;
            Bv[4 * q + 2] = t4.z; Bv[4 * q + 3] = t4.w;
            const float4 c4 = Cp[q];
            Cv[4 * q] = c4.x; Cv[4 * q + 1] = c4.y;
            Cv[4 * q + 2] = c4.z; Cv[4 * q + 3] = c4.w;
        }
        float yv = uv * Dv;
        #pragma unroll
        for (int s = 0; s < DS; ++s) {
            const float dA = __expf(dtv * Ac[s]);
            h[s] = dA * h[s] + du * Bv[s];
            yv += h[s] * Cv[s];
        }
        yp[(size_t)i * DI] = yv * silu_f(zv);
    }
}

// -------------------------------------------------------------------------
// Kernel 5: out_proj (128 -> 64) + bias, scatter to (B,C,H,W) layout.
// Per lane the 8 accumulator rows are 8 consecutive n -> two b128 stores.
// -------------------------------------------------------------------------
__global__ __launch_bounds__(128) void k_outproj(
    const float* __restrict__ y, const float* __restrict__ opw,
    const float* __restrict__ opb, float* __restrict__ out)
{
    __shared__ float ylds[64 * 132];
    const int tid = threadIdx.x;
    const int g0  = blockIdx.x * 64;

    for (int i = tid; i < 64 * DI; i += 128)
        ylds[(i >> 7) * 132 + (i & 127)] = y[(size_t)g0 * DI + i];
    __syncthreads();

    const int wave = tid >> 5, lane = tid & 31;
    const int mrow = lane & 15, hi = lane >> 4;
    const int khalf = hi * 2;
    const int bI = g0 >> 14;
    const int n0 = g0 & (Nn - 1);
    for (int i = 0; i < 4; ++i) {
        const int job = wave * 4 + i;    // 16 jobs: tt(0..3) x ot(0..3)
        const int tt = job >> 2, ot = job & 3;
        const float* arow = &ylds[(tt * 16 + mrow) * 132];
        const float* brow = opw + (ot * 16 + mrow) * 128;
        v8f acc = {0.f, 0.f, 0.f, 0.f, 0.f, 0.f, 0.f, 0.f};
        #pragma unroll
        for (int kk = 0; kk < 32; ++kk) {
            const int k0 = kk * 4 + khalf;
            v2f av = { arow[k0], arow[k0 + 1] };
            v2f bv = { brow[k0], brow[k0 + 1] };
            acc = wmma_f32(av, bv, acc);
        }
        const int c = ot * 16 + mrow;
        const float bias = opb[c];
        float* op = out + (size_t)bI * NC * Nn + (size_t)c * Nn
                        + (n0 + tt * 16 + hi * 8);
        const float4 lo = make_float4(acc[0] + bias, acc[1] + bias,
                                      acc[2] + bias, acc[3] + bias);
        const float4 hi4 = make_float4(acc[4] + bias, acc[5] + bias,
                                       acc[6] + bias, acc[7] + bias);
        *reinterpret_cast<float4*>(op)     = lo;
        *reinterpret_cast<float4*>(op + 4) = hi4;
    }
}

// -------------------------------------------------------------------------
extern "C" void kernel_launch(void* const* d_in, const int* in_sizes, int n_in,
                              void* d_out, int out_size, void* d_ws, size_t ws_size,
                              hipStream_t stream)
{
    (void)in_sizes; (void)n_in; (void)out_size; (void)ws_size;
    const float* x         = (const float*)d_in[0];
    const float* ln_w      = (const float*)d_in[1];
    const float* ln_b      = (const float*)d_in[2];
    const float* in_proj_w = (const float*)d_in[3];
    const float* conv_w    = (const float*)d_in[4];
    const float* conv_b    = (const float*)d_in[5];
    const float* x_proj_w  = (const float*)d_in[6];
    const float* dt_proj_w = (const float*)d_in[7];
    const float* dt_proj_b = (const float*)d_in[8];
    const float* A_log     = (const float*)d_in[9];
    const float* D_param   = (const float*)d_in[10];
    const float* out_pw    = (const float*)d_in[11];
    const float* out_pb    = (const float*)d_in[12];

    float* ws = (float*)d_ws;
    size_t o = 0;
    float* xin = ws + o; o += (size_t)NT * DI;
    float* zb  = ws + o; o += (size_t)NT * DI;
    float* u   = ws + o; o += (size_t)NT * DI;
    float* dtv = ws + o; o += (size_t)NT * DI;
    float* yv  = ws + o; o += (size_t)NT * DI;
    float* Bc  = ws + o; o += (size_t)NT * DS;
    float* Cc  = ws + o; o += (size_t)NT * DS;
    const size_t S = (size_t)Bb * NCH * DI * DS;
    float* pA  = ws + o; o += S;
    float* hf  = ws + o; o += S;
    float* cr  = ws + o; o += S;

    k_ln_inproj <<<NT / 64, 256, 0, stream>>>(x, ln_w, ln_b, in_proj_w, xin, zb);
    k_conv_silu <<<(NT * DI) / 256, 256, 0, stream>>>(xin, conv_w, conv_b, u);
    k_xproj_dt  <<<NT / 64, 128, 0, stream>>>(u, x_proj_w, dt_proj_w, dt_proj_b,
                                              dtv, Bc, Cc);
    k_scan_pass1<<<Bb * NCH, 128, 0, stream>>>(dtv, u, Bc, A_log, pA, hf);
    k_scan_carry<<<(Bb * DI * DS) / 256, 256, 0, stream>>>(pA, hf, cr);
    k_scan_pass2<<<Bb * NCH, 128, 0, stream>>>(dtv, u, zb, Bc, Cc, A_log,
                                               D_param, cr, yv);
    k_outproj   <<<NT / 64, 128, 0, stream>>>(yv, out_pw, out_pb, (float*)d_out);
}